// Attention_67774583931211
// MI455X (gfx1250) — compile-verified
//
#include <hip/hip_runtime.h>
#include <math.h>

// Luong 'general' attention decode step, restructured for MI455X:
//   score[b,s] = enc[b,s,:] . (state[b] @ Wa)   (kills the 137-GFLOP GEMM)
//   single-pass online softmax + context over enc (256 MB read once ~11us @ 23.3 TB/s)
//   two small f32 GEMMs done with V_WMMA_F32_16X16X4_F32.

typedef __attribute__((ext_vector_type(2))) float v2f;
typedef __attribute__((ext_vector_type(8))) float v8f;

#define B_DIM 32
#define S_DIM 2048
#define H_DIM 1024
#define SPLIT 8
#define ROWS_PER_BLOCK (S_DIM / SPLIT)     // 256 rows per workgroup
#define ROWS_PER_WAVE  (ROWS_PER_BLOCK/8)  // 32 rows per wave (8 waves/WG)

// workspace layout (float element offsets; total ~426k floats = 1.7 MB)
#define WS_U      0                              // [B,H]       u = state @ Wa
#define WS_SCORES (WS_U + B_DIM*H_DIM)           // [B,S]       raw scores
#define WS_PCTX   (WS_SCORES + B_DIM*S_DIM)      // [B,SPLIT,H] partial contexts
#define WS_PM     (WS_PCTX + B_DIM*SPLIT*H_DIM)  // [B,SPLIT]   partial maxes
#define WS_PL     (WS_PM + B_DIM*SPLIT)          // [B,SPLIT]   partial sums
#define WS_TANH   (WS_PL + B_DIM*SPLIT)          // [B,2H]      tanh(concat)

// ---------------------------------------------------------------------------
// Kernel A: u[b,h] = sum_g state[b,g] * Wa[g,h]   via V_WMMA_F32_16X16X4_F32
// grid (H/16, B/16), 1 wave per block. A = state tile [16b x 4g], B = Wa [4g x 16h].
// ---------------------------------------------------------------------------
__global__ __launch_bounds__(32)
void k_state_proj(const float* __restrict__ state, const float* __restrict__ Wa,
                  float* __restrict__ u) {
  const int lane = threadIdx.x & 31;
  const int n    = lane & 15;            // N index (h) / M index (b) for A
  const int koff = (lane >> 4) * 2;      // which K pair this lane-half holds
  const int h0   = blockIdx.x * 16;
  const int b0   = blockIdx.y * 16;

  v8f c = {};
  for (int k = 0; k < H_DIM; k += 4) {
    // A 16x4 f32: lane m=lane%16 holds K = koff, koff+1 (contiguous in state row)
    v2f a = *(const v2f*)(state + (size_t)(b0 + n) * H_DIM + k + koff);
    // B 4x16 f32: lane n holds B[koff..koff+1][n] = Wa[(k+koff..)*H + h0+n]
    v2f b;
    b.x = Wa[(size_t)(k + koff)     * H_DIM + h0 + n];
    b.y = Wa[(size_t)(k + koff + 1) * H_DIM + h0 + n];
    c = __builtin_amdgcn_wmma_f32_16x16x4_f32(false, a, false, b, (short)0, c,
                                              false, false);
  }
#pragma unroll
  for (int r = 0; r < 8; ++r) {
    int m = r + 8 * (lane >> 4);         // C/D layout: VGPR r, lane-half selects M+8
    u[(size_t)(b0 + m) * H_DIM + h0 + n] = c[r];
  }
}

// ---------------------------------------------------------------------------
// Kernel B: fused scores + online-softmax context, one pass over enc.
// grid (SPLIT, B), 256 threads (8 waves). Each wave: 32 rows; row lives in
// registers (8x float4); butterfly wave32 reduce gives the score to all lanes;
// running (m,l,ctx) updated with rescaling. Per-WG partials -> workspace.
// ---------------------------------------------------------------------------
__global__ __launch_bounds__(256)
void k_scores_ctx(const float* __restrict__ enc, const float* __restrict__ u,
                  float* __restrict__ scores, float* __restrict__ pctx,
                  float* __restrict__ pm, float* __restrict__ pl) {
  __shared__ float s_ctx[8 * H_DIM];   // 32 KB: per-wave context vectors
  __shared__ float s_m[8];
  __shared__ float s_l[8];

  const int chunk = blockIdx.x;
  const int b     = blockIdx.y;
  const int w     = threadIdx.x >> 5;
  const int lane  = threadIdx.x & 31;

  // u[b,:] resident in registers: lane covers cols {j*128 + lane*4 .. +3}
  const float4* u4 = (const float4*)(u + (size_t)b * H_DIM);
  float4 uv[8];
#pragma unroll
  for (int j = 0; j < 8; ++j) uv[j] = u4[j * 32 + lane];

  float m = -INFINITY, l = 0.0f;
  float4 acc[8];
#pragma unroll
  for (int j = 0; j < 8; ++j) acc[j] = make_float4(0.f, 0.f, 0.f, 0.f);

  const int row0 = chunk * ROWS_PER_BLOCK + w * ROWS_PER_WAVE;
  for (int i = 0; i < ROWS_PER_WAVE; ++i) {
    const int s = row0 + i;
    const float4* row4 = (const float4*)(enc + ((size_t)b * S_DIM + s) * H_DIM);
    float4 rv[8];
#pragma unroll
    for (int j = 0; j < 8; ++j) rv[j] = row4[j * 32 + lane];   // global_load_b128

    float dot = 0.f;
#pragma unroll
    for (int j = 0; j < 8; ++j)
      dot += rv[j].x * uv[j].x + rv[j].y * uv[j].y +
             rv[j].z * uv[j].z + rv[j].w * uv[j].w;
#pragma unroll
    for (int off = 16; off >= 1; off >>= 1)
      dot += __shfl_xor(dot, off, 32);   // all 32 lanes get the full score

    if (lane == 0) scores[(size_t)b * S_DIM + s] = dot;

    // online softmax update (uniform across wave)
    float mn    = fmaxf(m, dot);
    float scale = __expf(m - mn);
    float p     = __expf(dot - mn);
    l = l * scale + p;
#pragma unroll
    for (int j = 0; j < 8; ++j) {
      acc[j].x = acc[j].x * scale + p * rv[j].x;
      acc[j].y = acc[j].y * scale + p * rv[j].y;
      acc[j].z = acc[j].z * scale + p * rv[j].z;
      acc[j].w = acc[j].w * scale + p * rv[j].w;
    }
    m = mn;
  }

  // per-wave partials -> LDS
  float4* sc4 = (float4*)s_ctx;
#pragma unroll
  for (int j = 0; j < 8; ++j) sc4[w * 256 + j * 32 + lane] = acc[j];
  if (lane == 0) { s_m[w] = m; s_l[w] = l; }
  __syncthreads();

  // combine 8 waves; thread tid owns 4 columns
  const int tid = threadIdx.x;
  float mb = s_m[0];
#pragma unroll
  for (int ww = 1; ww < 8; ++ww) mb = fmaxf(mb, s_m[ww]);
  float lb = 0.f;
  float4 cb = make_float4(0.f, 0.f, 0.f, 0.f);
#pragma unroll
  for (int ww = 0; ww < 8; ++ww) {
    float sc = __expf(s_m[ww] - mb);
    lb += sc * s_l[ww];
    float4 v = sc4[ww * 256 + tid];
    cb.x += sc * v.x; cb.y += sc * v.y; cb.z += sc * v.z; cb.w += sc * v.w;
  }
  ((float4*)pctx)[((size_t)b * SPLIT + chunk) * 256 + tid] = cb;
  if (tid == 0) { pm[b * SPLIT + chunk] = mb; pl[b * SPLIT + chunk] = lb; }
}

// ---------------------------------------------------------------------------
// Kernel C: combine SPLIT partials per batch; write softmax_attn and
// tanh(concat(context, state)) for the output GEMM. grid (B), 256 threads.
// ---------------------------------------------------------------------------
__global__ __launch_bounds__(256)
void k_combine(const float* __restrict__ state, const float* __restrict__ scores,
               const float* __restrict__ pctx, const float* __restrict__ pm,
               const float* __restrict__ pl, float* __restrict__ tb,
               float* __restrict__ attn_out) {
  const int b   = blockIdx.x;
  const int tid = threadIdx.x;

  float mg = pm[b * SPLIT];
#pragma unroll
  for (int w = 1; w < SPLIT; ++w) mg = fmaxf(mg, pm[b * SPLIT + w]);
  float lg = 0.f;
  float4 ctx = make_float4(0.f, 0.f, 0.f, 0.f);
#pragma unroll
  for (int w = 0; w < SPLIT; ++w) {
    float sc = __expf(pm[b * SPLIT + w] - mg);
    lg += sc * pl[b * SPLIT + w];
    float4 v = ((const float4*)pctx)[((size_t)b * SPLIT + w) * 256 + tid];
    ctx.x += sc * v.x; ctx.y += sc * v.y; ctx.z += sc * v.z; ctx.w += sc * v.w;
  }
  const float inv = 1.0f / lg;
  ctx.x *= inv; ctx.y *= inv; ctx.z *= inv; ctx.w *= inv;

  // tanh(concat): [0,H) = tanh(context), [H,2H) = tanh(state)
  float* trow = tb + (size_t)b * (2 * H_DIM);
  const int h = tid * 4;
  trow[h + 0] = tanhf(ctx.x);
  trow[h + 1] = tanhf(ctx.y);
  trow[h + 2] = tanhf(ctx.z);
  trow[h + 3] = tanhf(ctx.w);
  float4 sv = ((const float4*)(state + (size_t)b * H_DIM))[tid];
  trow[H_DIM + h + 0] = tanhf(sv.x);
  trow[H_DIM + h + 1] = tanhf(sv.y);
  trow[H_DIM + h + 2] = tanhf(sv.z);
  trow[H_DIM + h + 3] = tanhf(sv.w);

  // softmax_attn[b,:]
  for (int s = tid; s < S_DIM; s += 256)
    attn_out[(size_t)b * S_DIM + s] =
        __expf(scores[(size_t)b * S_DIM + s] - mg) * inv;
}

// ---------------------------------------------------------------------------
// Kernel D: out[b,n] = relu( sum_k tb[b,k]*Wc[n,k] + bias[n] ) via WMMA f32.
// grid (H/16, B/16), 1 wave per block. K = 2H = 2048.
// ---------------------------------------------------------------------------
__global__ __launch_bounds__(32)
void k_out_gemm(const float* __restrict__ tb, const float* __restrict__ Wc,
                const float* __restrict__ bias, float* __restrict__ out) {
  const int lane = threadIdx.x & 31;
  const int n    = lane & 15;
  const int koff = (lane >> 4) * 2;
  const int h0   = blockIdx.x * 16;   // output feature tile
  const int b0   = blockIdx.y * 16;   // batch tile

  v8f c = {};
  for (int k = 0; k < 2 * H_DIM; k += 4) {
    v2f a  = *(const v2f*)(tb + (size_t)(b0 + n) * (2 * H_DIM) + k + koff);
    v2f bb = *(const v2f*)(Wc + (size_t)(h0 + n) * (2 * H_DIM) + k + koff);
    c = __builtin_amdgcn_wmma_f32_16x16x4_f32(false, a, false, bb, (short)0, c,
                                              false, false);
  }
  const float bv = bias[h0 + n];
#pragma unroll
  for (int r = 0; r < 8; ++r) {
    int m = r + 8 * (lane >> 4);
    out[(size_t)(b0 + m) * H_DIM + h0 + n] = fmaxf(c[r] + bv, 0.0f);
  }
}

// ---------------------------------------------------------------------------
extern "C" void kernel_launch(void* const* d_in, const int* in_sizes, int n_in,
                              void* d_out, int out_size, void* d_ws, size_t ws_size,
                              hipStream_t stream) {
  const float* enc   = (const float*)d_in[0];  // [B,S,H]
  const float* state = (const float*)d_in[1];  // [1,B,H] -> [B,H]
  const float* Wa    = (const float*)d_in[2];  // [H,H]
  const float* Wc    = (const float*)d_in[3];  // [H,2H]
  const float* Wcb   = (const float*)d_in[4];  // [H]

  float* out  = (float*)d_out;                 // [0,32768): output  [32768,98304): attn
  float* ws   = (float*)d_ws;
  float* u      = ws + WS_U;
  float* scores = ws + WS_SCORES;
  float* pctx   = ws + WS_PCTX;
  float* pm     = ws + WS_PM;
  float* pl     = ws + WS_PL;
  float* tbuf   = ws + WS_TANH;

  k_state_proj<<<dim3(H_DIM / 16, B_DIM / 16), 32, 0, stream>>>(state, Wa, u);
  k_scores_ctx<<<dim3(SPLIT, B_DIM), 256, 0, stream>>>(enc, u, scores, pctx, pm, pl);
  k_combine<<<dim3(B_DIM), 256, 0, stream>>>(state, scores, pctx, pm, pl, tbuf,
                                             out + B_DIM * H_DIM);
  k_out_gemm<<<dim3(H_DIM / 16, B_DIM / 16), 32, 0, stream>>>(tbuf, Wc, Wcb, out);
}